// EnhancedGNNModel_65283502899679
// MI455X (gfx1250) — compile-verified
//
#include <hip/hip_runtime.h>
#include <hip/hip_bf16.h>

// GNN: 4x SAGEConv(HID=64) + BN + ReLU, then 5 SAGE heads (3,2,5,9,6 dims).
// N=1e6 nodes, E=5e6 edges. Memory-bound on edge aggregation; dense parts via
// V_WMMA_F32_16X16X4_F32 (f32 WMMA keeps reference numerics; GEMM cost trivial).
// Weights stored in LDS pre-paired (k, k+1) as float2 so each WMMA B operand is
// one aligned ds_load_b64 into an even VGPR pair (no repack moves).

typedef float v2f __attribute__((ext_vector_type(2)));
typedef float v8f __attribute__((ext_vector_type(8)));

#define NN 1000000
#define NE 5000000
#define HID 64
#define BN_EPS 1e-5f

// ---------------------------------------------------------------- zero fill
__global__ void k_zero_f4(float4* p, long n4) {
    long i = (long)blockIdx.x * blockDim.x + threadIdx.x;
    long stride = (long)gridDim.x * blockDim.x;
    float4 z = make_float4(0.f, 0.f, 0.f, 0.f);
    for (; i < n4; i += stride) p[i] = z;
}

// ---------------------------------------------------------------- degree
__global__ void k_count(const int* __restrict__ dst, float* cnt, int E) {
    int i = blockIdx.x * blockDim.x + threadIdx.x;
    if (i < E) atomicAdd(&cnt[dst[i]], 1.0f);
}

// ------------------------------------------------- conv1 scatter (scalar x)
__global__ void k_scatter_x(const int* __restrict__ src, const int* __restrict__ dst,
                            const float* __restrict__ x, float* agg, int E) {
    int i = blockIdx.x * blockDim.x + threadIdx.x;
    if (i < E) atomicAdd(&agg[dst[i]], x[src[i]]);
}

// ---------------------------------------- hidden scatter: one edge per wave
// lane L moves channels 2L, 2L+1 as a float2 (coalesced 256B load per wave).
// Prefetch the source row 64 edges ahead (global_prefetch_b8) to hide HBM
// latency on the random gather stream.
__global__ void k_scatter_h(const int* __restrict__ src, const int* __restrict__ dst,
                            const float* __restrict__ h, float* agg, int E) {
    int w = (int)(((long)blockIdx.x * blockDim.x + threadIdx.x) >> 5);
    int lane = threadIdx.x & 31;
    if (w >= E) return;
    int wn = w + 64;
    if (wn < E) {
        int sn = src[wn];
        __builtin_prefetch(h + (long)sn * HID + lane * 2, 0, 0);
    }
    int s = src[w];
    int d = dst[w];
    float2 v = *(const float2*)(h + (long)s * HID + lane * 2);
    float* p = agg + (long)d * HID + lane * 2;
    atomicAdd(p, v.x);
    atomicAdd(p + 1, v.y);
}

// --------------------------------------------------- conv1 (cin=1) + BN + ReLU
// thread t handles node t>>4, channels 4*(t&15)..+3 (coalesced float4 stores)
__global__ void k_conv1(const float* __restrict__ x, const float* __restrict__ aggS,
                        const float* __restrict__ cnt,
                        const float* __restrict__ Wl, const float* __restrict__ Wr,
                        const float* __restrict__ bb,
                        const float* __restrict__ gm, const float* __restrict__ bt,
                        const float* __restrict__ rm, const float* __restrict__ rv,
                        float* __restrict__ h) {
    int t = blockIdx.x * blockDim.x + threadIdx.x;
    int n = t >> 4;
    int c0 = (t & 15) << 2;
    if (n >= NN) return;
    float xi = x[n];
    float m = aggS[n] / fmaxf(cnt[n], 1.0f);
    float4 o;
    float* po = (float*)&o;
#pragma unroll
    for (int j = 0; j < 4; ++j) {
        int c = c0 + j;
        float v = m * Wl[c] + xi * Wr[c] + bb[c];
        float sc = gm[c] * rsqrtf(rv[c] + BN_EPS);
        v = (v - rm[c]) * sc + bt[c];
        po[j] = fmaxf(v, 0.0f);
    }
    *(float4*)(h + (long)n * HID + c0) = o;
}

// ------------------------------------------------------- hidden SAGE layer
// out[16x64 tile] = [mean|h](16x128) @ [Wl;Wr](128x64), BN+bias folded, ReLU.
// One wave per 16-row tile; V_WMMA_F32_16X16X4_F32, 4 N-tiles, K=128 in steps
// of 4. A layout: lanes 0-15 = rows, k pair {kk,kk+1}; lanes 16-31 = {kk+2,kk+3}
// -> per-lane k base always even -> paired float2 weight fetch is exact.
// Writes in place over `agg` (rows fully consumed before epilogue).
__global__ __launch_bounds__(256) void k_sage_hidden(
    const float* agg, const float* __restrict__ h, const float* __restrict__ cnt,
    const float* __restrict__ Wl, const float* __restrict__ Wr,
    const float* __restrict__ bb,
    const float* __restrict__ gm, const float* __restrict__ bt,
    const float* __restrict__ rm, const float* __restrict__ rv,
    float* out) {
    __shared__ v2f Wp[64 * HID];   // pair p -> (Wcomb[2p][n], Wcomb[2p+1][n]); 32KB
    __shared__ float scaleS[HID];
    __shared__ float addS[HID];

    int t = threadIdx.x;
    for (int i = t; i < 64 * HID; i += 256) {
        int p = i >> 6;
        int n = i & 63;
        v2f v;
        if (p < 32) {              // combined k = 2p, 2p+1 in [0,64) -> Wl
            v.x = Wl[(2 * p) * HID + n];
            v.y = Wl[(2 * p + 1) * HID + n];
        } else {                   // combined k in [64,128) -> Wr
            int k = 2 * p - 64;
            v.x = Wr[k * HID + n];
            v.y = Wr[(k + 1) * HID + n];
        }
        Wp[i] = v;
    }
    if (t < HID) {
        float sc = gm[t] * rsqrtf(rv[t] + BN_EPS);
        scaleS[t] = sc;
        addS[t] = (bb[t] - rm[t]) * sc + bt[t];
    }
    __syncthreads();

    int wave = t >> 5;
    int lane = t & 31;
    int tile = blockIdx.x * 8 + wave;
    if (tile * 16 >= NN) return;   // whole-wave uniform; EXEC stays all-ones

    int r0 = tile * 16;
    int rowA = r0 + (lane & 15);
    int kph = (lane >> 4) << 1;      // 0 or 2
    int lg = lane >> 4;              // 0 or 1 (pair offset)
    int colb = lane & 15;
    float invc = 1.0f / fmaxf(cnt[rowA], 1.0f);
    const float* aggRow = agg + (long)rowA * HID;
    const float* hRow = h + (long)rowA * HID;

    v8f acc[4];
#pragma unroll
    for (int nt = 0; nt < 4; ++nt) acc[nt] = (v8f){0.f,0.f,0.f,0.f,0.f,0.f,0.f,0.f};

    // K = 0..63 : mean part (agg * 1/deg)
#pragma unroll 4
    for (int kk = 0; kk < 64; kk += 4) {
        int k0 = kk + kph;
        v2f a = *(const v2f*)(aggRow + k0);
        a.x *= invc;
        a.y *= invc;
        const v2f* wrow = &Wp[((kk >> 1) + lg) * HID];
#pragma unroll
        for (int nt = 0; nt < 4; ++nt) {
            v2f b = wrow[nt * 16 + colb];
            acc[nt] = __builtin_amdgcn_wmma_f32_16x16x4_f32(
                false, a, false, b, (short)0, acc[nt], false, false);
        }
    }
    // K = 64..127 : root part (h)
#pragma unroll 4
    for (int kk = 0; kk < 64; kk += 4) {
        int k0 = kk + kph;
        v2f a = *(const v2f*)(hRow + k0);
        const v2f* wrow = &Wp[(32 + (kk >> 1) + lg) * HID];
#pragma unroll
        for (int nt = 0; nt < 4; ++nt) {
            v2f b = wrow[nt * 16 + colb];
            acc[nt] = __builtin_amdgcn_wmma_f32_16x16x4_f32(
                false, a, false, b, (short)0, acc[nt], false, false);
        }
    }

    // epilogue: D layout -> VGPR r: lanes0-15 M=r, lanes16-31 M=r+8, N=lane&15
    int rowOff = lg * 8;
#pragma unroll
    for (int nt = 0; nt < 4; ++nt) {
        int col = nt * 16 + colb;
        float sc = scaleS[col];
        float ad = addS[col];
#pragma unroll
        for (int r = 0; r < 8; ++r) {
            float v = acc[nt][r] * sc + ad;
            out[(long)(r0 + rowOff + r) * HID + col] = fmaxf(v, 0.0f);
        }
    }
}

// ----------------------------------------------------------------- heads
struct HeadW {
    const float* wl[5];
    const float* wr[5];
    const float* b[5];
};

__device__ __forceinline__ void col_to_head(int c, int& hd, int& lc, int& dim) {
    if (c >= 19)      { hd = 4; lc = c - 19; dim = 6; }
    else if (c >= 10) { hd = 3; lc = c - 10; dim = 9; }
    else if (c >= 5)  { hd = 2; lc = c - 5;  dim = 5; }
    else if (c >= 3)  { hd = 1; lc = c - 3;  dim = 2; }
    else              { hd = 0; lc = c;      dim = 3; }
}

// All 5 heads fused: [mean|h](16x128) @ Wcat(128x32, cols 25..31 zero-padded)
__global__ __launch_bounds__(256) void k_sage_heads(
    const float* __restrict__ agg, const float* __restrict__ h,
    const float* __restrict__ cnt, HeadW HW, float* __restrict__ out) {
    __shared__ v2f Wp[64 * 32];    // pair p -> (Wcat[2p][c], Wcat[2p+1][c]); 16KB
    __shared__ float biasS[32];

    int t = threadIdx.x;
    for (int i = t; i < 64 * 32; i += 256) {
        int p = i >> 5, c = i & 31;
        v2f v = (v2f){0.f, 0.f};
        if (c < 25) {
            int hd, lc, dim;
            col_to_head(c, hd, lc, dim);
            if (p < 32) {
                v.x = HW.wl[hd][(2 * p) * dim + lc];
                v.y = HW.wl[hd][(2 * p + 1) * dim + lc];
            } else {
                int k = 2 * p - 64;
                v.x = HW.wr[hd][k * dim + lc];
                v.y = HW.wr[hd][(k + 1) * dim + lc];
            }
        }
        Wp[i] = v;
    }
    if (t < 32) {
        float v = 0.0f;
        if (t < 25) {
            int hd, lc, dim;
            col_to_head(t, hd, lc, dim);
            v = HW.b[hd][lc];
        }
        biasS[t] = v;
    }
    __syncthreads();

    int wave = t >> 5;
    int lane = t & 31;
    int tile = blockIdx.x * 8 + wave;
    if (tile * 16 >= NN) return;

    int r0 = tile * 16;
    int rowA = r0 + (lane & 15);
    int kph = (lane >> 4) << 1;
    int lg = lane >> 4;
    int colb = lane & 15;
    float invc = 1.0f / fmaxf(cnt[rowA], 1.0f);
    const float* aggRow = agg + (long)rowA * HID;
    const float* hRow = h + (long)rowA * HID;

    v8f acc0 = (v8f){0.f,0.f,0.f,0.f,0.f,0.f,0.f,0.f};
    v8f acc1 = acc0;

#pragma unroll 4
    for (int kk = 0; kk < 64; kk += 4) {
        int k0 = kk + kph;
        v2f a = *(const v2f*)(aggRow + k0);
        a.x *= invc;
        a.y *= invc;
        const v2f* wrow = &Wp[((kk >> 1) + lg) * 32];
        v2f b0 = wrow[colb];
        v2f b1 = wrow[16 + colb];
        acc0 = __builtin_amdgcn_wmma_f32_16x16x4_f32(false, a, false, b0, (short)0, acc0, false, false);
        acc1 = __builtin_amdgcn_wmma_f32_16x16x4_f32(false, a, false, b1, (short)0, acc1, false, false);
    }
#pragma unroll 4
    for (int kk = 0; kk < 64; kk += 4) {
        int k0 = kk + kph;
        v2f a = *(const v2f*)(hRow + k0);
        const v2f* wrow = &Wp[(32 + (kk >> 1) + lg) * 32];
        v2f b0 = wrow[colb];
        v2f b1 = wrow[16 + colb];
        acc0 = __builtin_amdgcn_wmma_f32_16x16x4_f32(false, a, false, b0, (short)0, acc0, false, false);
        acc1 = __builtin_amdgcn_wmma_f32_16x16x4_f32(false, a, false, b1, (short)0, acc1, false, false);
    }

    int rowOff = lg * 8;
#pragma unroll
    for (int nt = 0; nt < 2; ++nt) {
        int col = nt * 16 + colb;
        if (col < 25) {
            int hd, lc, dim;
            col_to_head(col, hd, lc, dim);
            long base = (hd == 0 ? 0L : hd == 1 ? 3L : hd == 2 ? 5L : hd == 3 ? 10L : 19L) * (long)NN;
            float bv = biasS[col];
            const v8f& a = nt ? acc1 : acc0;
#pragma unroll
            for (int r = 0; r < 8; ++r) {
                out[base + (long)(r0 + rowOff + r) * dim + lc] = a[r] + bv;
            }
        }
    }
}

// ----------------------------------------------------------------- driver
extern "C" void kernel_launch(void* const* d_in, const int* in_sizes, int n_in,
                              void* d_out, int out_size, void* d_ws, size_t ws_size,
                              hipStream_t stream) {
    const float* x = (const float*)d_in[0];
    const int* ei = (const int*)d_in[1];
    const int* src = ei;
    const int* dst = ei + NE;

    // params in setup_inputs() dict order (recursive insertion order)
    const float* cWl[4] = {(const float*)d_in[2], (const float*)d_in[5], (const float*)d_in[8],  (const float*)d_in[11]};
    const float* cWr[4] = {(const float*)d_in[3], (const float*)d_in[6], (const float*)d_in[9],  (const float*)d_in[12]};
    const float* cB[4]  = {(const float*)d_in[4], (const float*)d_in[7], (const float*)d_in[10], (const float*)d_in[13]};
    const float* bnG[4] = {(const float*)d_in[14], (const float*)d_in[18], (const float*)d_in[22], (const float*)d_in[26]};
    const float* bnB[4] = {(const float*)d_in[15], (const float*)d_in[19], (const float*)d_in[23], (const float*)d_in[27]};
    const float* bnM[4] = {(const float*)d_in[16], (const float*)d_in[20], (const float*)d_in[24], (const float*)d_in[28]};
    const float* bnV[4] = {(const float*)d_in[17], (const float*)d_in[21], (const float*)d_in[25], (const float*)d_in[29]};
    HeadW HW;
    for (int i = 0; i < 5; ++i) {
        HW.wl[i] = (const float*)d_in[30 + 3 * i];
        HW.wr[i] = (const float*)d_in[31 + 3 * i];
        HW.b[i]  = (const float*)d_in[32 + 3 * i];
    }

    // workspace: cnt (N) | bufA (N*64) | bufB (N*64)
    float* cnt = (float*)d_ws;
    float* bufA = cnt + NN;
    float* bufB = bufA + (size_t)NN * HID;

    const int ethreads = 256;
    const int eblocks = (NE + ethreads - 1) / ethreads;        // thread/edge
    const int ewblocks = (NE + 7) / 8;                         // wave/edge
    const int tileBlocks = ((NN / 16) + 7) / 8;                // wave/16-row tile

    // degree (once) + conv1 scalar aggregation
    k_zero_f4<<<1024, 256, 0, stream>>>((float4*)cnt, (long)NN / 4);
    k_zero_f4<<<1024, 256, 0, stream>>>((float4*)bufA, (long)NN / 4);
    k_count<<<eblocks, ethreads, 0, stream>>>(dst, cnt, NE);
    k_scatter_x<<<eblocks, ethreads, 0, stream>>>(src, dst, x, bufA, NE);
    k_conv1<<<NN / 16, 256, 0, stream>>>(x, bufA, cnt, cWl[0], cWr[0], cB[0],
                                         bnG[0], bnB[0], bnM[0], bnV[0], bufB);

    float* h = bufB;
    float* o = bufA;
    for (int L = 1; L <= 3; ++L) {   // conv2..conv4
        k_zero_f4<<<8192, 256, 0, stream>>>((float4*)o, (long)NN * HID / 4);
        k_scatter_h<<<ewblocks, 256, 0, stream>>>(src, dst, h, o, NE);
        k_sage_hidden<<<tileBlocks, 256, 0, stream>>>(o, h, cnt, cWl[L], cWr[L], cB[L],
                                                      bnG[L], bnB[L], bnM[L], bnV[L], o);
        float* tmp = h; h = o; o = tmp;
    }

    // heads: one shared aggregation of h4, fused 5-head WMMA transform
    k_zero_f4<<<8192, 256, 0, stream>>>((float4*)o, (long)NN * HID / 4);
    k_scatter_h<<<ewblocks, 256, 0, stream>>>(src, dst, h, o, NE);
    k_sage_heads<<<tileBlocks, 256, 0, stream>>>(o, h, cnt, HW, (float*)d_out);
}